// RotatedIoULoss_1537598292532
// MI455X (gfx1250) — compile-verified
//
#include <hip/hip_runtime.h>

#define REF_EPS 1e-8f

typedef __attribute__((ext_vector_type(2))) float v2f;
typedef __attribute__((ext_vector_type(8))) float v8f;

// Monotonic substitute for atan2 ordering over (-pi, pi].
__device__ __forceinline__ float pseudo_angle(float x, float y) {
    float p = x / (fabsf(x) + fabsf(y) + 1e-30f);   // in [-1, 1]
    return (y >= 0.0f) ? (1.0f - p) : (p - 1.0f);   // in [-2, 2], increasing with angle
}

__global__ __launch_bounds__(256) void riou_main_kernel(
    const float* __restrict__ pred,
    const float* __restrict__ target,
    const unsigned char* __restrict__ pos,   // jax bool => 1 byte
    float* __restrict__ iou_out,             // d_out + 1
    float* __restrict__ partials,            // 2 floats per block
    int n)
{
    const int tid  = threadIdx.x;
    const int gid  = blockIdx.x * 256 + tid;
    const bool inR = (gid < n);
    const int  ri  = inR ? gid : (n - 1);

    // ---- load boxes -------------------------------------------------------
    float b1[5], b2[5];
    #pragma unroll
    for (int k = 0; k < 5; ++k) { b1[k] = pred[ri * 5 + k]; b2[k] = target[ri * 5 + k]; }

    const float D2R = 0.017453292519943295f;
    const float ang1 = b1[4] * D2R;                          // deg2rad(pred_a)
    const float ang2 = (b2[4] * 180.0f - 180.0f) * D2R;      // deg2rad(t_a*180-180)

    // ---- corners ----------------------------------------------------------
    const float SX[4] = { 0.5f, -0.5f, -0.5f,  0.5f };
    const float SY[4] = { 0.5f,  0.5f, -0.5f, -0.5f };
    float c1x[4], c1y[4], c2x[4], c2y[4];
    {
        float sn = sinf(ang1), cs = cosf(ang1);
        #pragma unroll
        for (int j = 0; j < 4; ++j) {
            float xs = SX[j] * b1[2], ys = SY[j] * b1[3];
            c1x[j] = xs * cs - ys * sn + b1[0];
            c1y[j] = xs * sn + ys * cs + b1[1];
        }
        sn = sinf(ang2); cs = cosf(ang2);
        #pragma unroll
        for (int j = 0; j < 4; ++j) {
            float xs = SX[j] * b2[2], ys = SY[j] * b2[3];
            c2x[j] = xs * cs - ys * sn + b2[0];
            c2y[j] = xs * sn + ys * cs + b2[1];
        }
    }

    // ---- candidate vertex set (24 real + 8 pad) ---------------------------
    float vx[32], vy[32];
    bool  vm[24];

    #pragma unroll
    for (int j = 0; j < 4; ++j) { vx[j]     = c1x[j]; vy[j]     = c1y[j]; }
    #pragma unroll
    for (int j = 0; j < 4; ++j) { vx[4 + j] = c2x[j]; vy[4 + j] = c2y[j]; }

    const float tol = 1e-6f;
    // c1 corners inside box2
    {
        float ax = c2x[0], ay = c2y[0];
        float abx = c2x[1] - ax, aby = c2y[1] - ay;
        float adx = c2x[3] - ax, ady = c2y[3] - ay;
        float iab = 1.0f / (abx * abx + aby * aby + REF_EPS);
        float iad = 1.0f / (adx * adx + ady * ady + REF_EPS);
        #pragma unroll
        for (int j = 0; j < 4; ++j) {
            float amx = c1x[j] - ax, amy = c1y[j] - ay;
            float pab = (abx * amx + aby * amy) * iab;
            float pad = (adx * amx + ady * amy) * iad;
            vm[j] = (pab > -tol) & (pab < 1.0f + tol) & (pad > -tol) & (pad < 1.0f + tol);
        }
    }
    // c2 corners inside box1
    {
        float ax = c1x[0], ay = c1y[0];
        float abx = c1x[1] - ax, aby = c1y[1] - ay;
        float adx = c1x[3] - ax, ady = c1y[3] - ay;
        float iab = 1.0f / (abx * abx + aby * aby + REF_EPS);
        float iad = 1.0f / (adx * adx + ady * ady + REF_EPS);
        #pragma unroll
        for (int j = 0; j < 4; ++j) {
            float amx = c2x[j] - ax, amy = c2y[j] - ay;
            float pab = (abx * amx + aby * amy) * iab;
            float pad = (adx * amx + ady * amy) * iad;
            vm[4 + j] = (pab > -tol) & (pab < 1.0f + tol) & (pad > -tol) & (pad < 1.0f + tol);
        }
    }
    // 16 edge-pair intersections
    #pragma unroll
    for (int i1 = 0; i1 < 4; ++i1) {
        float px = c1x[i1], py = c1y[i1];
        float rx = c1x[(i1 + 1) & 3] - px, ry = c1y[(i1 + 1) & 3] - py;
        #pragma unroll
        for (int i2 = 0; i2 < 4; ++i2) {
            float qx = c2x[i2], qy = c2y[i2];
            float sx = c2x[(i2 + 1) & 3] - qx, sy = c2y[(i2 + 1) & 3] - qy;
            float qpx = qx - px, qpy = qy - py;
            float rxs = rx * sy - ry * sx;
            float inv = 1.0f / (rxs + REF_EPS);
            float t = (qpx * sy - qpy * sx) * inv;
            float u = (qpx * ry - qpy * rx) * inv;
            bool mk = (t > 0.0f) & (t < 1.0f) & (u > 0.0f) & (u < 1.0f);
            int id = 8 + i1 * 4 + i2;
            vm[id] = mk;
            vx[id] = mk ? (px + t * rx) : 0.0f;
            vy[id] = mk ? (py + t * ry) : 0.0f;
        }
    }

    // ---- centroid of masked verts ----------------------------------------
    float nv = 0.0f, mxs = 0.0f, mys = 0.0f;
    #pragma unroll
    for (int j = 0; j < 24; ++j) {
        float mf = vm[j] ? 1.0f : 0.0f;
        nv  += mf;
        mxs += vx[j] * mf;
        mys += vy[j] * mf;
    }
    float invnv = 1.0f / fmaxf(nv, 1.0f);
    float mx = mxs * invnv, my = mys * invnv;

    // ---- sort keys (pseudo-angle; masked-out => 1e9; pads => 2e9) --------
    float key[32];
    #pragma unroll
    for (int j = 0; j < 24; ++j) {
        float dx = vx[j] - mx, dy = vy[j] - my;
        vx[j] = dx; vy[j] = dy;
        key[j] = vm[j] ? pseudo_angle(dx, dy) : 1e9f;
    }
    #pragma unroll
    for (int j = 24; j < 32; ++j) { key[j] = 2e9f; vx[j] = 0.0f; vy[j] = 0.0f; }

    // ---- 32-wide bitonic sorting network (fully unrolled, register-resident)
    #pragma unroll
    for (int k = 2; k <= 32; k <<= 1) {
        #pragma unroll
        for (int j = k >> 1; j > 0; j >>= 1) {
            #pragma unroll
            for (int i = 0; i < 32; ++i) {
                int l = i ^ j;
                if (l > i) {
                    bool up = ((i & k) == 0);
                    float ka = key[i], kb = key[l];
                    bool sw = up ? (ka > kb) : (ka < kb);
                    float xa = vx[i], xb = vx[l];
                    float ya = vy[i], yb = vy[l];
                    key[i] = sw ? kb : ka; key[l] = sw ? ka : kb;
                    vx[i]  = sw ? xb : xa; vx[l]  = sw ? xa : xb;
                    vy[i]  = sw ? yb : ya; vy[l]  = sw ? ya : yb;
                }
            }
        }
    }

    // ---- replace unmasked with first sorted vertex, shoelace over 24 -----
    const float MASK_THRESH = 5e8f;
    float fx = vx[0], fy = vy[0];
    #pragma unroll
    for (int j = 0; j < 24; ++j) {
        bool mj = key[j] < MASK_THRESH;
        vx[j] = mj ? vx[j] : fx;
        vy[j] = mj ? vy[j] : fy;
    }
    float cr = 0.0f;
    #pragma unroll
    for (int j = 0; j < 24; ++j) {
        int jn = (j + 1) % 24;
        cr += vx[j] * vy[jn] - vy[j] * vx[jn];
    }
    float inter = 0.5f * fabsf(cr);
    inter = (nv >= 3.0f) ? inter : 0.0f;

    // ---- IoU --------------------------------------------------------------
    float area1 = b1[2] * b1[3];
    float area2 = b2[2] * b2[3];
    float uni = area1 + area2 - inter;
    float iou = inter / (uni + REF_EPS);
    iou = fminf(fmaxf(iou, 1e-7f), 1.0f - 1e-7f);

    if (inR) iou_out[gid] = iou;

    float mf = (inR && pos[ri] != 0) ? 1.0f : 0.0f;
    float la = (1.0f - iou) * mf;   // (1 - iou) * mask
    float lm = mf;                  // mask count

    // ---- wave reduction via the matrix pipe -------------------------------
    // B (4x16 f32): VGPR0 = la per lane, VGPR1 = lm per lane.
    // A rows: M=0 -> (1,1,0,0), M=1 -> (0,0,1,1); per A layout that means
    // A-VGPR0 = A-VGPR1 = 1.0 in lanes {0,17}, else 0.
    // D[0][N] = la_N + la_{N+16};  D[1][N] = lm_N + lm_{N+16}  (lanes 0..15)
    int lane = tid & 31;
    float aval = (lane == 0 || lane == 17) ? 1.0f : 0.0f;
    v2f A; A.x = aval; A.y = aval;
    v2f B; B.x = la;   B.y = lm;
    v8f C = {0.f, 0.f, 0.f, 0.f, 0.f, 0.f, 0.f, 0.f};
    v8f D = __builtin_amdgcn_wmma_f32_16x16x4_f32(
        false, A, false, B, (short)0, C, false, false);
    float ra = D[0];   // la pair-sums, valid lanes 0..15 (rows M>=2 are zero)
    float rm = D[1];   // lm pair-sums

    #pragma unroll
    for (int off = 8; off >= 1; off >>= 1) {
        ra += __shfl_xor(ra, off, 32);
        rm += __shfl_xor(rm, off, 32);
    }

    __shared__ float sA[8], sM[8];
    int wid = tid >> 5;
    if (lane == 0) { sA[wid] = ra; sM[wid] = rm; }
    __syncthreads();
    if (tid == 0) {
        float ta = 0.0f, tm = 0.0f;
        #pragma unroll
        for (int w = 0; w < 8; ++w) { ta += sA[w]; tm += sM[w]; }
        partials[2 * blockIdx.x]     = ta;
        partials[2 * blockIdx.x + 1] = tm;
    }
}

__global__ __launch_bounds__(256) void riou_reduce_kernel(
    const float* __restrict__ partials, int nblocks, float* __restrict__ out)
{
    __shared__ float sa[256], sm[256];
    int tid = threadIdx.x;
    float a = 0.0f, m = 0.0f;
    for (int i = tid; i < nblocks; i += 256) {
        a += partials[2 * i];
        m += partials[2 * i + 1];
    }
    sa[tid] = a; sm[tid] = m;
    __syncthreads();
    #pragma unroll
    for (int s = 128; s > 0; s >>= 1) {
        if (tid < s) { sa[tid] += sa[tid + s]; sm[tid] += sm[tid + s]; }
        __syncthreads();
    }
    if (tid == 0) out[0] = sa[0] / fmaxf(sm[0], 1.0f);
}

extern "C" void kernel_launch(void* const* d_in, const int* in_sizes, int n_in,
                              void* d_out, int out_size, void* d_ws, size_t ws_size,
                              hipStream_t stream) {
    const float* pred   = (const float*)d_in[0];
    const float* target = (const float*)d_in[1];
    const unsigned char* pos = (const unsigned char*)d_in[2];  // jax bool = 1B
    float* out      = (float*)d_out;        // out[0]=loss, out[1..N]=ious
    float* partials = (float*)d_ws;

    int n = in_sizes[0] / 5;
    int blocks = (n + 255) / 256;

    riou_main_kernel<<<blocks, 256, 0, stream>>>(pred, target, pos,
                                                 out + 1, partials, n);
    riou_reduce_kernel<<<1, 256, 0, stream>>>(partials, blocks, out);
}